// RankingLoss_48730698940765
// MI455X (gfx1250) — compile-verified
//
#include <hip/hip_runtime.h>
#include <math.h>
#include <stdint.h>

// ---------------------------------------------------------------------------
// RankingLoss: loss = mean_b sum_v relu(0.1 + (x_b . e_v) - (x_b . e_{t_b}))
// x = l2-normalized inputs.  B=1024, V=128000, D=512, fp32 throughout.
// GEMM on V_WMMA_F32_16X16X4_F32; global->LDS staging via CDNA5
// GLOBAL_LOAD_ASYNC_TO_LDS_B128 with double-buffered LDS tiles.
// ---------------------------------------------------------------------------

typedef __attribute__((ext_vector_type(2))) float v2f;
typedef __attribute__((ext_vector_type(8))) float v8f;

constexpr int   kB = 1024;
constexpr int   kV = 128000;
constexpr int   kD = 512;
constexpr float kMargin = 0.1f;

constexpr int   BM = 64;        // batch rows per block
constexpr int   BN = 128;       // vocab rows per block
constexpr int   BK = 32;        // K chunk staged in LDS
constexpr int   LDSW = BK + 4;  // padded row stride (36 floats): conflict-free
constexpr int   NTHREADS = 256; // 8 wave32s
constexpr int   NCHUNK = kD / BK;

__global__ void RankingLoss_zero_out(float* out) { out[0] = 0.0f; }

// One block per batch row: sumsq + positive-class dot in one pass,
// then write normalized row and neg[b] = (row . emb[target]) / max(||row||,eps)
__global__ __launch_bounds__(NTHREADS)
void RankingLoss_normalize(const float* __restrict__ in,
                           const float* __restrict__ emb,
                           const int*   __restrict__ tgt,
                           float* __restrict__ xn,
                           float* __restrict__ neg) {
  __shared__ float s_sq[NTHREADS];
  __shared__ float s_dot[NTHREADS];
  __shared__ float s_inv;
  const int b = blockIdx.x;
  const int tid = threadIdx.x;
  const float* row = in + (size_t)b * kD;
  const float* er  = emb + (size_t)tgt[b] * kD;

  float sq = 0.0f, dt = 0.0f;
  for (int k = tid; k < kD; k += NTHREADS) {
    float v = row[k];
    sq += v * v;
    dt += v * er[k];
  }
  s_sq[tid] = sq;
  s_dot[tid] = dt;
  __syncthreads();
  for (int off = NTHREADS / 2; off > 0; off >>= 1) {
    if (tid < off) {
      s_sq[tid]  += s_sq[tid + off];
      s_dot[tid] += s_dot[tid + off];
    }
    __syncthreads();
  }
  if (tid == 0) {
    float n = fmaxf(sqrtf(s_sq[0]), 1e-12f);
    float inv = 1.0f / n;
    s_inv = inv;
    neg[b] = s_dot[0] * inv;
  }
  __syncthreads();
  const float inv = s_inv;
  for (int k = tid; k < kD; k += NTHREADS) {
    xn[(size_t)b * kD + k] = row[k] * inv;
  }
}

// CDNA5 async copy: 16B global -> LDS, no VGPR staging, tracked by ASYNCcnt.
// LDS destination VGPR carries the workgroup-relative byte offset (low 32
// bits of the generic shared-memory pointer, per ISA flat-LDS aperture).
__device__ __forceinline__ void async_b128(float* lds_dst, const float* gsrc) {
  uint32_t lds_off = (uint32_t)(uintptr_t)lds_dst;
  asm volatile("global_load_async_to_lds_b128 %0, %1, off"
               :: "v"(lds_off), "v"(gsrc)
               : "memory");
}

__device__ __forceinline__ void wait_async0() {
  asm volatile("s_wait_asynccnt 0x0" ::: "memory");
}

// Issue one K-chunk's worth of async copies: 6 x B128 per thread.
__device__ __forceinline__ void stage_async(float* __restrict__ Asb,
                                            float* __restrict__ Bsb,
                                            const float* __restrict__ xn,
                                            const float* __restrict__ emb,
                                            int kk, int tid, int Mbase, int Nbase) {
#pragma unroll
  for (int t = 0; t < (BM * BK) / 4 / NTHREADS; ++t) {   // 2 per thread
    int i = tid + t * NTHREADS;
    int r = i >> 3, c = (i & 7) << 2;
    async_b128(&Asb[r * LDSW + c], &xn[(size_t)(Mbase + r) * kD + kk + c]);
  }
#pragma unroll
  for (int t = 0; t < (BN * BK) / 4 / NTHREADS; ++t) {   // 4 per thread
    int i = tid + t * NTHREADS;
    int r = i >> 3, c = (i & 7) << 2;
    async_b128(&Bsb[r * LDSW + c], &emb[(size_t)(Nbase + r) * kD + kk + c]);
  }
}

// Tiled fp32 WMMA GEMM fused with margin-ranking epilogue.
// Grid: (kV/BN, kB/BM). Wave w owns M-subtile (w>>1) and N-subtiles
// 4*(w&1) .. 4*(w&1)+3  (4 accumulators of 16x16 each).
__global__ __launch_bounds__(NTHREADS)
void RankingLoss_wmma(const float* __restrict__ xn,
                      const float* __restrict__ emb,
                      const float* __restrict__ neg,
                      float* __restrict__ out) {
  __shared__ float As[2][BM * LDSW];   // 2 x  9,216 B
  __shared__ float Bs[2][BN * LDSW];   // 2 x 18,432 B
  __shared__ float red[NTHREADS];

  const int tid  = threadIdx.x;
  const int lane = tid & 31;
  const int wid  = tid >> 5;
  const int Mbase = blockIdx.y * BM;
  const int Nbase = blockIdx.x * BN;
  const int mi = wid >> 1;          // 0..3   : M subtile
  const int nb = (wid & 1) * 4;     // 0 or 4 : first N subtile

  v8f acc[4] = {};

  // A-matrix 16x4 fp32 ISA layout: lanes 0-15 hold K={k,k+1} of row M=lane,
  // lanes 16-31 hold K={k+2,k+3} of row M=lane-16.  B 4x16 mirrors it
  // (lanes 0-15: first K half, lanes 16-31: second K half, N = lane&15).
  const int koff = (lane >> 4) << 1;
  const int l15  = lane & 15;
  const int arow = mi * 16 + l15;

  // Prologue: async-stage chunk 0 into buffer 0.
  stage_async(As[0], Bs[0], xn, emb, 0, tid, Mbase, Nbase);
  wait_async0();
  __syncthreads();

  for (int c = 0; c < NCHUNK; ++c) {
    const int cur = c & 1;
    // Overlap: issue next chunk's async copies into the other buffer while
    // this chunk's WMMAs run.  (Other buffer was last read at iteration c-1,
    // sealed by that iteration's barrier -> no WAR hazard.)
    if (c + 1 < NCHUNK) {
      stage_async(As[cur ^ 1], Bs[cur ^ 1], xn, emb, (c + 1) * BK,
                  tid, Mbase, Nbase);
    }

    const float* Asb = As[cur];
    const float* Bsb = Bs[cur];
#pragma unroll
    for (int k4 = 0; k4 < BK; k4 += 4) {
      v2f a = *(const v2f*)&Asb[arow * LDSW + k4 + koff];
#pragma unroll
      for (int j = 0; j < 4; ++j) {
        int brow = (nb + j) * 16 + l15;
        v2f bv = *(const v2f*)&Bsb[brow * LDSW + k4 + koff];
        acc[j] = __builtin_amdgcn_wmma_f32_16x16x4_f32(
            /*neg_a=*/false, a, /*neg_b=*/false, bv,
            /*c_mod=*/(short)0, acc[j], /*reuse_a=*/false, /*reuse_b=*/false);
      }
    }

    wait_async0();     // my async writes for chunk c+1 have landed in LDS
    __syncthreads();   // all waves done computing c and staging c+1
  }

  // ---- fused loss epilogue ----
  // C/D layout: VGPR r, lanes 0-15 -> M=r, lanes 16-31 -> M=r+8; N = lane&15.
  float sum = 0.0f;
  const int mrow0 = Mbase + mi * 16 + ((lane >> 4) << 3);
#pragma unroll
  for (int r = 0; r < 8; ++r) {
    float nv = neg[mrow0 + r];
#pragma unroll
    for (int j = 0; j < 4; ++j) {
      sum += fmaxf(0.0f, kMargin + acc[j][r] - nv);
    }
  }
  red[tid] = sum;
  __syncthreads();
  for (int off = NTHREADS / 2; off > 0; off >>= 1) {
    if (tid < off) red[tid] += red[tid + off];
    __syncthreads();
  }
  if (tid == 0) atomicAdd(out, red[0] * (1.0f / (float)kB));
}

extern "C" void kernel_launch(void* const* d_in, const int* in_sizes, int n_in,
                              void* d_out, int out_size, void* d_ws, size_t ws_size,
                              hipStream_t stream) {
  (void)in_sizes; (void)n_in; (void)out_size; (void)ws_size;
  const float* inputs = (const float*)d_in[0];   // [B, D] fp32
  const float* emb    = (const float*)d_in[1];   // [V, D] fp32
  const int*   target = (const int*)d_in[2];     // [B]   int32
  float* out = (float*)d_out;                    // scalar fp32

  float* xn  = (float*)d_ws;                     // [B, D] normalized inputs
  float* neg = xn + (size_t)kB * kD;             // [B] positive scores

  RankingLoss_zero_out<<<1, 1, 0, stream>>>(out);
  RankingLoss_normalize<<<kB, NTHREADS, 0, stream>>>(inputs, emb, target, xn, neg);
  dim3 grid(kV / BN, kB / BM);                   // (1000, 16)
  RankingLoss_wmma<<<grid, NTHREADS, 0, stream>>>(xn, emb, neg, out);
}